// Projector_21157008900255
// MI455X (gfx1250) — compile-verified
//
#include <hip/hip_runtime.h>
#include <stdint.h>

// L2-normalize + cosine-vs-8-prototypes + softmax(K) + attention-weighted
// projection, fused into a single memory-bound pass.
// inputs: [B, 200, 64] f32;  prototype: [8, 64] f32;  out: [B, 8, 64] f32.
//
// CDNA5 paths used:
//  - global_load_async_to_lds_b128 (ASYNCcnt) for the dominant 210MB staging
//  - v_wmma_f32_16x16x4_f32 for the attn @ X projection GEMM

typedef __attribute__((ext_vector_type(2))) float v2f;
typedef __attribute__((ext_vector_type(8))) float v8f;

#define LDIM 200
#define DDIM 64
#define KDIM 8
#define LPAD 208          // L padded to a multiple of 4 (WMMA k-step)
#define XS   68           // LDS row stride for X (floats): 16B-aligned, bank-skewed

__global__ __launch_bounds__(128)
void projector_kernel(const float* __restrict__ inputs,
                      const float* __restrict__ proto,
                      float* __restrict__ out)
{
    __shared__ float Xs[LPAD * XS];       // staged inputs[b], zero-padded rows
    __shared__ float Pn[KDIM * DDIM];     // L2-normalized prototypes
    __shared__ float attnT[LPAD * 16];    // attn^T: [l][k], k padded 8->16 with 0

    const int tid = threadIdx.x;
    const int b   = blockIdx.x;
    const float* __restrict__ xb = inputs + (size_t)b * (LDIM * DDIM);

    // ---- Phase 1a: async-stage inputs[b] into LDS, bypassing VGPRs.
    // Each lane issues 25 GLOBAL_LOAD_ASYNC_TO_LDS_B128 (ASYNCcnt <= 25 < 63).
    for (int j = tid; j < (LDIM * DDIM) / 4; j += 128) {
        const int l = j >> 4;             // 16 float4 per 64-float row
        const int c = (j & 15) << 2;
        // Generic LDS pointer = {aperture_hi, lds_offset}; low 32 bits = AS3 offset.
        const uint32_t lds_off = (uint32_t)(uintptr_t)(&Xs[l * XS + c]);
        const void*    gaddr   = (const void*)(xb + (size_t)j * 4);
        asm volatile("global_load_async_to_lds_b128 %0, %1, off"
                     :: "v"(lds_off), "v"(gaddr)
                     : "memory");
    }

    // ---- Phase 0: normalize prototypes into LDS (tiny, L2-resident reads);
    // overlaps with the async staging above.
    if (tid < KDIM) {
        const float* p = proto + tid * DDIM;
        float ss = 0.f;
        #pragma unroll 8
        for (int d = 0; d < DDIM; ++d) { float t = p[d]; ss += t * t; }
        const float rn = rsqrtf(fmaxf(ss, 1e-12f));
        #pragma unroll 8
        for (int d = 0; d < DDIM; ++d) Pn[tid * DDIM + d] = p[d] * rn;
    }

    // ---- Phase 1b: zero the pad rows (regular DS stores, disjoint from async)
    {
        const float4 z = make_float4(0.f, 0.f, 0.f, 0.f);
        for (int j = tid; j < (LPAD - LDIM) * 16; j += 128) {
            const int l = LDIM + (j >> 4);
            const int c = (j & 15) << 2;
            *(float4*)&Xs[l * XS + c] = z;
        }
    }

    // Drain async global->LDS copies, then make them visible to all waves.
    asm volatile("s_wait_asynccnt 0x0" ::: "memory");
    __syncthreads();

    // ---- Phase 2/3: per-l norm, cosine vs 8 prototypes, softmax over k
    for (int l = tid; l < LPAD; l += 128) {
        if (l < LDIM) {
            float acc[KDIM];
            #pragma unroll
            for (int k = 0; k < KDIM; ++k) acc[k] = 0.f;
            float ss = 0.f;
            #pragma unroll 4
            for (int d = 0; d < DDIM; d += 4) {
                const float4 x = *(const float4*)&Xs[l * XS + d];
                ss += x.x * x.x + x.y * x.y + x.z * x.z + x.w * x.w;
                #pragma unroll
                for (int k = 0; k < KDIM; ++k) {
                    const float* pr = &Pn[k * DDIM + d];   // broadcast reads
                    acc[k] += x.x * pr[0] + x.y * pr[1] + x.z * pr[2] + x.w * pr[3];
                }
            }
            const float rn = rsqrtf(fmaxf(ss, 1e-12f));
            float m = -3.0e38f;
            #pragma unroll
            for (int k = 0; k < KDIM; ++k) { acc[k] *= rn; m = fmaxf(m, acc[k]); }
            float s = 0.f;
            #pragma unroll
            for (int k = 0; k < KDIM; ++k) { const float e = __expf(acc[k] - m); acc[k] = e; s += e; }
            const float inv = 1.0f / s;
            #pragma unroll
            for (int k = 0; k < KDIM; ++k) attnT[l * 16 + k] = acc[k] * inv;
            #pragma unroll
            for (int k = KDIM; k < 16; ++k) attnT[l * 16 + k] = 0.f;
        } else {
            #pragma unroll
            for (int k = 0; k < 16; ++k) attnT[l * 16 + k] = 0.f;  // pad rows
        }
    }
    __syncthreads();

    // ---- Phase 4: out[b, 0:8, dtile] = attn(16x208) @ X(208x64) via
    //      V_WMMA_F32_16X16X4_F32; each wave owns one 16-wide D tile.
    const int wave  = tid >> 5;
    const int lane  = tid & 31;
    const int half  = lane >> 4;        // K-pair select per ISA A/B layout
    const int lm    = lane & 15;        // M (A) / N (B) index
    const int dbase = wave * 16;

    v8f c = {0.f, 0.f, 0.f, 0.f, 0.f, 0.f, 0.f, 0.f};
    #pragma unroll 4
    for (int kk = 0; kk < LPAD; kk += 4) {
        const int r0 = kk + 2 * half;
        v2f a, bb;
        a[0]  = attnT[r0 * 16 + lm];            // A[m=lm, k=r0-kk]
        a[1]  = attnT[(r0 + 1) * 16 + lm];      // A[m=lm, k=r0-kk+1]
        bb[0] = Xs[r0 * XS + dbase + lm];       // B[k, n=lm]
        bb[1] = Xs[(r0 + 1) * XS + dbase + lm];
        c = __builtin_amdgcn_wmma_f32_16x16x4_f32(
                /*neg_a=*/false, a, /*neg_b=*/false, bb,
                /*c_mod=*/(short)0, c, /*reuse_a=*/false, /*reuse_b=*/false);
    }

    // C/D layout: lanes 0-15 hold M=0..7 in VGPR 0..7 (exactly k=0..7), N=lane.
    if (half == 0) {
        float* __restrict__ ob = out + (size_t)b * (KDIM * DDIM) + dbase + lm;
        #pragma unroll
        for (int k = 0; k < KDIM; ++k) ob[k * DDIM] = c[k];
    }
}

extern "C" void kernel_launch(void* const* d_in, const int* in_sizes, int n_in,
                              void* d_out, int out_size, void* d_ws, size_t ws_size,
                              hipStream_t stream) {
    const float* inputs = (const float*)d_in[0];
    const float* proto  = (const float*)d_in[1];
    float* out          = (float*)d_out;
    const int B = in_sizes[0] / (LDIM * DDIM);   // 4096 for the reference shapes
    projector_kernel<<<dim3(B), dim3(128), 0, stream>>>(inputs, proto, out);
}